// Transcoder_36764920054294
// MI455X (gfx1250) — compile-verified
//
#include <hip/hip_runtime.h>
#include <hip/hip_bf16.h>

#define BATCH   8192
#define IN_DIM  256
#define HID_DIM 16384
#define OUT_DIM 256
#define TOPK    256
#define KC      256   // decoder K-chunk staged in LDS

typedef __attribute__((ext_vector_type(2))) float    v2f;
typedef __attribute__((ext_vector_type(8))) float    v8f;
typedef __attribute__((ext_vector_type(4))) unsigned v4u;
typedef __attribute__((ext_vector_type(4))) int      v4i;
typedef __attribute__((ext_vector_type(8))) int      v8i;

#if defined(__has_builtin)
#if __has_builtin(__builtin_amdgcn_tensor_load_to_lds) && \
    __has_builtin(__builtin_amdgcn_s_wait_tensorcnt)
#define HAVE_TDM 1
#endif
#endif
#ifndef HAVE_TDM
#define HAVE_TDM 0
#endif

static __device__ __forceinline__ v8f wmma4(v2f a, v2f b, v8f c) {
    // D = A(16x4 f32) * B(4x16 f32) + C(16x16 f32)
    return __builtin_amdgcn_wmma_f32_16x16x4_f32(
        false, a, false, b, (short)0, c, false, false);
}

#if HAVE_TDM
// Issue one 2D TDM tile load: 32 rows x KC floats from `gptr` (row stride
// HID_DIM floats) into LDS at byte offset `lds_addr`.  D# layout per CDNA5
// ISA "Tensor DMA Descriptor" groups 0/1; groups 2/3 zero (2D tile).
static __device__ __forceinline__ void tdm_load_tile(unsigned lds_addr,
                                                     const float* gptr) {
    unsigned long long ga = (unsigned long long)(uintptr_t)gptr;
    v4u g0;
    g0[0] = 1u;                                   // count=1, user descriptor
    g0[1] = lds_addr;                             // lds_addr (bytes)
    g0[2] = (unsigned)ga;                         // global_addr[31:0]
    g0[3] = (unsigned)((ga >> 32) & 0x01FFFFFFu)  // global_addr[56:32]
          | (2u << 30);                           // type = 2 ("image")
    v8i g1;
    g1[0] = (int)(2u << 16);                      // wg_mask=0, data_size=4B
    g1[1] = (int)((unsigned)HID_DIM << 16);       // tensor_dim0[15:0] @ [31:16]
    g1[2] = (int)((unsigned)BATCH  << 16);        // tensor_dim1[15:0] @ [31:16]
    g1[3] = (int)((unsigned)KC     << 16);        // tile_dim0 @ [31:16]
    g1[4] = 32;                                   // tile_dim1 (tile_dim2=0)
    g1[5] = HID_DIM;                              // tensor_dim0_stride[31:0]
    g1[6] = 0;                                    // stride0 hi / stride1 lo
    g1[7] = 0;
    v4i gz = {0, 0, 0, 0};
#if __clang_major__ >= 23
    v8i gz8 = {0, 0, 0, 0, 0, 0, 0, 0};
    __builtin_amdgcn_tensor_load_to_lds(g0, g1, gz, gz, gz8, 0);
#else
    __builtin_amdgcn_tensor_load_to_lds(g0, g1, gz, gz, 0);
#endif
}
#endif

// ---------------------------------------------------------------------------
// Kernel 1: hidden = relu(x @ enc_w.T + enc_b)
// block = 256 threads (8 waves). Block tile: M=32, N=256.
// ---------------------------------------------------------------------------
__global__ __launch_bounds__(256) void sae_enc_kernel(
    const float* __restrict__ x, const float* __restrict__ enc_w,
    const float* __restrict__ enc_b, float* __restrict__ hidden)
{
    const int lane = threadIdx.x & 31;
    const int wave = threadIdx.x >> 5;
    const int r    = lane & 15;
    const int kh   = (lane >> 4) << 1;
    const int m0   = blockIdx.x * 32 + (wave >> 2) * 16;
    const int n0   = blockIdx.y * 256 + (wave & 3) * 64;

    const float* ap = x     + (size_t)(m0 + r) * IN_DIM + kh;
    const float* bp = enc_w + (size_t)(n0 + r) * IN_DIM + kh;

    v8f c0 = {}, c1 = {}, c2 = {}, c3 = {};
#pragma unroll 4
    for (int k = 0; k < IN_DIM; k += 4) {
        v2f a  = *(const v2f*)(ap + k);
        v2f b0 = *(const v2f*)(bp + k);
        v2f b1 = *(const v2f*)(bp + (size_t)16 * IN_DIM + k);
        v2f b2 = *(const v2f*)(bp + (size_t)32 * IN_DIM + k);
        v2f b3 = *(const v2f*)(bp + (size_t)48 * IN_DIM + k);
        c0 = wmma4(a, b0, c0);
        c1 = wmma4(a, b1, c1);
        c2 = wmma4(a, b2, c2);
        c3 = wmma4(a, b3, c3);
    }

    const int mrow = m0 + ((lane >> 4) << 3);
    v8f acc[4] = {c0, c1, c2, c3};
#pragma unroll
    for (int t = 0; t < 4; ++t) {
        const int col   = n0 + t * 16 + r;
        const float bia = enc_b[col];
        float* hp = hidden + (size_t)mrow * HID_DIM + col;
#pragma unroll
        for (int j = 0; j < 8; ++j) {
            float v = acc[t][j] + bia;
            hp[(size_t)j * HID_DIM] = v > 0.f ? v : 0.f;
        }
    }
}

// ---------------------------------------------------------------------------
// Kernel 2: exact k-th largest per row via 4-pass 8-bit radix select.
// ---------------------------------------------------------------------------
__global__ __launch_bounds__(256) void sae_topk_kernel(
    const float* __restrict__ hidden, float* __restrict__ thresh)
{
    __shared__ unsigned hist[256];
    __shared__ unsigned s_target, s_krem;
    const int tid = threadIdx.x;
    const float4* src = (const float4*)(hidden + (size_t)blockIdx.x * HID_DIM);

    float vals[64];
#pragma unroll
    for (int j = 0; j < 16; ++j) {
        float4 t = src[tid + j * 256];
        vals[4 * j + 0] = t.x; vals[4 * j + 1] = t.y;
        vals[4 * j + 2] = t.z; vals[4 * j + 3] = t.w;
    }

    unsigned prefix = 0u, mask = 0u, krem = TOPK;
    for (int pass = 0; pass < 4; ++pass) {
        const int shift = 24 - 8 * pass;
        hist[tid] = 0u;
        __syncthreads();
#pragma unroll
        for (int j = 0; j < 64; ++j) {
            unsigned u = __float_as_uint(vals[j]);
            if ((u & mask) == prefix)
                atomicAdd(&hist[(u >> shift) & 255u], 1u);
        }
        __syncthreads();
        if (tid == 0) {
            unsigned c = 0; int b = 255;
            for (; b >= 0; --b) {
                unsigned h = hist[b];
                if (c + h >= krem) break;
                c += h;
            }
            if (b < 0) b = 0;
            s_target = (unsigned)b;
            s_krem   = krem - c;
        }
        __syncthreads();
        prefix |= s_target << shift;
        mask   |= 255u << shift;
        krem    = s_krem;
        __syncthreads();
    }
    if (tid == 0) thresh[blockIdx.x] = __uint_as_float(prefix);
}

// ---------------------------------------------------------------------------
// Kernel 3: hidden_post = h * (h >= thr); out = hidden_post @ dec_w.T + dec_b
// Block tile: M=32 rows, N=256, K streamed in KC chunks.
// A-tile staged by the Tensor Data Mover into double-buffered LDS;
// wave 0 issues/waits (TENSORcnt), workgroup barrier releases consumers.
// ---------------------------------------------------------------------------
__global__ __launch_bounds__(256) void sae_dec_kernel(
    const float* __restrict__ dec_w, const float* __restrict__ dec_b,
    const float* __restrict__ thresh, float* __restrict__ hidden,
    float* __restrict__ out)
{
    __shared__ __align__(16) float atile[2 * 32 * KC];   // 2 x 32 KB
    const int tid  = threadIdx.x;
    const int lane = tid & 31;
    const int wave = tid >> 5;
    const int r    = lane & 15;
    const int kh   = (lane >> 4) << 1;
    const int m0   = blockIdx.x * 32;
    const int mh   = (wave >> 2) * 16;
    const int n0   = (wave & 3) * 64;

    const int tr = tid >> 3;                    // writeback row 0..31
    const int tc = tid & 7;                     // 8 threads per row
    const float thr_s = thresh[m0 + tr];
    const float thr_a = thresh[m0 + mh + r];

    const float* hbase = hidden + (size_t)m0 * HID_DIM;
    const int NCHUNK = HID_DIM / KC;            // 64

    v8f c0 = {}, c1 = {}, c2 = {}, c3 = {};

#if HAVE_TDM
    const unsigned lds_base = (unsigned)(uintptr_t)atile;  // flat[31:0] = LDS offset
    if (wave == 0)
        tdm_load_tile(lds_base, hbase);         // prefetch chunk 0 -> buf 0
#endif

    for (int ci = 0; ci < NCHUNK; ++ci) {
        float* curbuf = atile + (ci & 1) * (32 * KC);
        const int kc = ci * KC;

#if HAVE_TDM
        if (wave == 0) {
            if (ci + 1 < NCHUNK) {
                tdm_load_tile(lds_base + ((ci + 1) & 1) * (32 * KC * 4),
                              hbase + (size_t)(ci + 1) * KC);
                __builtin_amdgcn_s_wait_tensorcnt(1);  // chunk ci landed
            } else {
                __builtin_amdgcn_s_wait_tensorcnt(0);
            }
        }
        __syncthreads();
#else
        {   // synchronous staging fallback
            const float* grow = hbase + (size_t)tr * HID_DIM + kc;
            float* lrow = curbuf + tr * KC;
#pragma unroll
            for (int j = 0; j < 8; ++j) {
                int c4 = tc + j * 8;
                ((float4*)lrow)[c4] = ((const float4*)grow)[c4];
            }
            __syncthreads();
        }
#endif
        // write thresholded hidden_post back to global from the LDS stage
        {
            float* grow = hidden + (size_t)(m0 + tr) * HID_DIM + kc;
            const float* lrow = curbuf + tr * KC;
#pragma unroll
            for (int j = 0; j < 8; ++j) {
                int c4 = tc + j * 8;
                float4 v = ((const float4*)lrow)[c4];
                v.x = v.x >= thr_s ? v.x : 0.f;
                v.y = v.y >= thr_s ? v.y : 0.f;
                v.z = v.z >= thr_s ? v.z : 0.f;
                v.w = v.w >= thr_s ? v.w : 0.f;
                ((float4*)grow)[c4] = v;
            }
        }
        // WMMA over the chunk; threshold applied on A-fragment load
        const float* arow = curbuf + (mh + r) * KC + kh;
        const float* bp   = dec_w + (size_t)(n0 + r) * HID_DIM + kc + kh;
#pragma unroll 4
        for (int k = 0; k < KC; k += 4) {
            v2f a = *(const v2f*)(arow + k);
            a[0] = a[0] >= thr_a ? a[0] : 0.f;
            a[1] = a[1] >= thr_a ? a[1] : 0.f;
            v2f b0 = *(const v2f*)(bp + k);
            v2f b1 = *(const v2f*)(bp + (size_t)16 * HID_DIM + k);
            v2f b2 = *(const v2f*)(bp + (size_t)32 * HID_DIM + k);
            v2f b3 = *(const v2f*)(bp + (size_t)48 * HID_DIM + k);
            c0 = wmma4(a, b0, c0);
            c1 = wmma4(a, b1, c1);
            c2 = wmma4(a, b2, c2);
            c3 = wmma4(a, b3, c3);
        }
        __syncthreads();
    }

    const int mrow = m0 + mh + ((lane >> 4) << 3);
    v8f acc[4] = {c0, c1, c2, c3};
#pragma unroll
    for (int t = 0; t < 4; ++t) {
        const int col   = n0 + t * 16 + r;
        const float bia = dec_b[col];
        float* op = out + (size_t)mrow * OUT_DIM + col;
#pragma unroll
        for (int j = 0; j < 8; ++j)
            op[(size_t)j * OUT_DIM] = acc[t][j] + bia;
    }
}

// ---------------------------------------------------------------------------
extern "C" void kernel_launch(void* const* d_in, const int* in_sizes, int n_in,
                              void* d_out, int out_size, void* d_ws, size_t ws_size,
                              hipStream_t stream) {
    const float* x     = (const float*)d_in[0];
    const float* enc_w = (const float*)d_in[1];
    const float* enc_b = (const float*)d_in[2];
    const float* dec_w = (const float*)d_in[3];
    const float* dec_b = (const float*)d_in[4];
    // d_in[5] = k (fixed 256, baked into TOPK)

    float* out    = (float*)d_out;
    float* hidden = out + (size_t)BATCH * OUT_DIM;   // hidden_post output slot
    float* thresh = (float*)d_ws;                    // 8192 floats

    dim3 g1(BATCH / 32, HID_DIM / 256);
    sae_enc_kernel<<<g1, 256, 0, stream>>>(x, enc_w, enc_b, hidden);
    sae_topk_kernel<<<BATCH, 256, 0, stream>>>(hidden, thresh);
    sae_dec_kernel<<<BATCH / 32, 256, 0, stream>>>(dec_w, dec_b, thresh, hidden, out);
}